// KalmanFilter_20873541059116
// MI455X (gfx1250) — compile-verified
//
#include <hip/hip_runtime.h>

// Kalman filter, B=128, T=256, N=64, F=H=I, Q/R diagonal.
// Key structure: P/K recurrence is batch-independent -> hoist the 64x64
// inverse chain out of the batch dim (reference redundantly inverts 128
// identical matrices per step). Phase 1 computes K_t (t=0..255) with
// Gauss-Jordan (partial pivoting) + fp32 WMMA GEMMs; Phase 2 runs the
// per-batch scan X <- X + (Z - X) K^T with fp32 WMMA.

#define NDIM 64
#define TSTEPS 256
#define BATCH 128

typedef __attribute__((ext_vector_type(2))) float v2f;
typedef __attribute__((ext_vector_type(8))) float v8f;

// D(16x16,f32) = A(16x4,f32) x B(4x16,f32) + C  -- V_WMMA_F32_16X16X4_F32
__device__ inline v8f wmma_f32_16x16x4(v2f a, v2f b, v8f c) {
  return __builtin_amdgcn_wmma_f32_16x16x4_f32(
      /*neg_a=*/false, a, /*neg_b=*/false, b,
      /*c_mod=*/(short)0, c, /*reuse_a=*/false, /*reuse_b=*/false);
}

// ---------------------------------------------------------------------------
// Phase 1: single workgroup (16 waves = 512 threads) walks the covariance
// chain sequentially and emits K_t to global workspace.
//   P_prior = P + diag(Q)
//   S       = P_prior + diag(R)
//   Sinv    = GaussJordan(S)          (partial pivoting, matches linalg.inv)
//   K       = P_prior @ Sinv          (WMMA f32)
//   P_new   = P_prior - K @ P_prior   (WMMA f32)
// ---------------------------------------------------------------------------
__global__ __launch_bounds__(512)
void kalman_gain_kernel(const float* __restrict__ Q, const float* __restrict__ R,
                        float* __restrict__ Kout) {
  __shared__ float Pp[NDIM][NDIM];       // 16 KB: P (posterior) then P_prior
  __shared__ float AUG[NDIM][2 * NDIM];  // 32 KB: [S | I] -> [K | Sinv/Pnew]
  __shared__ float fac[NDIM];
  __shared__ float s_scal;
  __shared__ int s_piv;

  const int tid  = threadIdx.x;      // 0..511
  const int lane = tid & 31;
  const int wave = tid >> 5;         // 0..15 -> 4x4 grid of 16x16 tiles
  const int hl   = lane >> 4;        // half-wave select (0/1)
  const int ln   = lane & 15;
  const int tm   = (wave >> 2) * 16; // output tile row base
  const int tn   = (wave & 3) * 16;  // output tile col base

  // P0 = ones (torch.ones, NOT identity)
  for (int i = tid; i < NDIM * NDIM; i += 512) Pp[i / NDIM][i % NDIM] = 1.0f;
  __syncthreads();

  for (int t = 0; t < TSTEPS; ++t) {
    // P_prior = P + diag(Q)
    if (tid < NDIM) Pp[tid][tid] += Q[tid];
    __syncthreads();

    // AUG = [ P_prior + diag(R) | I ]
    for (int i = tid; i < NDIM * 2 * NDIM; i += 512) {
      int r = i / (2 * NDIM), c = i % (2 * NDIM);
      float v;
      if (c < NDIM) v = Pp[r][c] + (c == r ? R[r] : 0.0f);
      else          v = ((c - NDIM) == r) ? 1.0f : 0.0f;
      AUG[r][c] = v;
    }
    __syncthreads();

    // Gauss-Jordan with partial pivoting: AUG[:, N:2N] <- S^{-1}
    for (int k = 0; k < NDIM; ++k) {
      if (tid == 0) {
        int piv = k; float best = fabsf(AUG[k][k]);
        for (int r = k + 1; r < NDIM; ++r) {
          float v = fabsf(AUG[r][k]);
          if (v > best) { best = v; piv = r; }
        }
        s_piv = piv;
      }
      __syncthreads();
      const int piv = s_piv;
      if (piv != k && tid < 2 * NDIM) {
        float tmp = AUG[k][tid];
        AUG[k][tid] = AUG[piv][tid];
        AUG[piv][tid] = tmp;
      }
      __syncthreads();
      if (tid == 0) s_scal = 1.0f / AUG[k][k];
      __syncthreads();
      if (tid < 2 * NDIM) AUG[k][tid] *= s_scal;
      __syncthreads();
      if (tid < NDIM) fac[tid] = (tid == k) ? 0.0f : AUG[tid][k];
      __syncthreads();
      for (int i = tid; i < NDIM * 2 * NDIM; i += 512) {
        int r = i / (2 * NDIM), c = i % (2 * NDIM);
        AUG[r][c] -= fac[r] * AUG[k][c];  // fac[k]==0 keeps pivot row intact
      }
      __syncthreads();
    }

    // GEMM1: K = P_prior @ Sinv.  Reads Pp + AUG-right, writes AUG-left + global.
    {
      v8f acc = {0, 0, 0, 0, 0, 0, 0, 0};
      for (int kk = 0; kk < NDIM; kk += 4) {
        v2f a, b;
        a[0] = Pp[tm + ln][kk + 2 * hl + 0];
        a[1] = Pp[tm + ln][kk + 2 * hl + 1];
        b[0] = AUG[kk + 2 * hl + 0][NDIM + tn + ln];
        b[1] = AUG[kk + 2 * hl + 1][NDIM + tn + ln];
        acc = wmma_f32_16x16x4(a, b, acc);
      }
      float* Kt = Kout + (size_t)t * NDIM * NDIM;
      for (int r = 0; r < 8; ++r) {
        const int m = tm + r + 8 * hl;
        const int n = tn + ln;
        AUG[m][n] = acc[r];
        Kt[m * NDIM + n] = acc[r];
      }
    }
    __syncthreads();

    // GEMM2: P_new = P_prior + (-K) @ P_prior.  Reads AUG-left + Pp,
    // writes AUG-right (Sinv is dead now).
    {
      v8f acc;
      for (int r = 0; r < 8; ++r) acc[r] = Pp[tm + r + 8 * hl][tn + ln];
      for (int kk = 0; kk < NDIM; kk += 4) {
        v2f a, b;
        a[0] = -AUG[tm + ln][kk + 2 * hl + 0];
        a[1] = -AUG[tm + ln][kk + 2 * hl + 1];
        b[0] = Pp[kk + 2 * hl + 0][tn + ln];
        b[1] = Pp[kk + 2 * hl + 1][tn + ln];
        acc = wmma_f32_16x16x4(a, b, acc);
      }
      for (int r = 0; r < 8; ++r)
        AUG[tm + r + 8 * hl][NDIM + tn + ln] = acc[r];
    }
    __syncthreads();

    // P <- P_new
    for (int i = tid; i < NDIM * NDIM; i += 512)
      Pp[i / NDIM][i % NDIM] = AUG[i / NDIM][NDIM + i % NDIM];
    __syncthreads();
  }
}

// ---------------------------------------------------------------------------
// Phase 2: per-batch state scan. 8 workgroups x 16 batches; 4 waves each own
// one 16x16 (batch x state) tile. Per step:
//   X <- X + (Z_t - X) @ K_t^T   (WMMA f32, C-operand carries +X)
// K_t streamed from workspace (4 MB -> L2 resident at 192 MB).
// ---------------------------------------------------------------------------
__global__ __launch_bounds__(128)
void kalman_scan_kernel(const float* __restrict__ arr,
                        const float* __restrict__ Kmat,
                        float* __restrict__ out) {
  __shared__ float Xs[16][NDIM];  // current state, 16 batches
  __shared__ float As[16][NDIM];  // innovation Z - X

  const int tid  = threadIdx.x;   // 0..127
  const int lane = tid & 31;
  const int wave = tid >> 5;      // 0..3
  const int hl   = lane >> 4;
  const int ln   = lane & 15;
  const int i0   = wave * 16;     // state-dim tile base
  const int b0   = blockIdx.x * 16;

  for (int i = tid; i < 16 * NDIM; i += 128) Xs[i / NDIM][i % NDIM] = 0.0f;
  __syncthreads();

  for (int t = 0; t < TSTEPS; ++t) {
    // innovation
    for (int i = tid; i < 16 * NDIM; i += 128) {
      const int b = i / NDIM, j = i % NDIM;
      As[b][j] = arr[(size_t)(b0 + b) * TSTEPS * NDIM + (size_t)t * NDIM + j]
                 - Xs[b][j];
    }
    __syncthreads();

    // D = As @ K^T + X : B-fragment b[v] = K[i0+ln][kk+2*hl+v]
    v8f acc;
    for (int r = 0; r < 8; ++r) acc[r] = Xs[r + 8 * hl][i0 + ln];
    const float* Kt = Kmat + (size_t)t * NDIM * NDIM;
    for (int kk = 0; kk < NDIM; kk += 4) {
      v2f a, b;
      a[0] = As[ln][kk + 2 * hl + 0];
      a[1] = As[ln][kk + 2 * hl + 1];
      b[0] = Kt[(i0 + ln) * NDIM + kk + 2 * hl + 0];
      b[1] = Kt[(i0 + ln) * NDIM + kk + 2 * hl + 1];
      acc = wmma_f32_16x16x4(a, b, acc);
    }
    __syncthreads();  // all reads of Xs for this step are complete

    for (int r = 0; r < 8; ++r) {
      const int b = r + 8 * hl;
      Xs[b][i0 + ln] = acc[r];
      out[(size_t)(b0 + b) * TSTEPS * NDIM + (size_t)t * NDIM + i0 + ln] = acc[r];
    }
    __syncthreads();
  }
}

extern "C" void kernel_launch(void* const* d_in, const int* in_sizes, int n_in,
                              void* d_out, int out_size, void* d_ws, size_t ws_size,
                              hipStream_t stream) {
  const float* arr = (const float*)d_in[0];  // [128,256,64] fp32
  const float* Q   = (const float*)d_in[1];  // [64,1] fp32
  const float* R   = (const float*)d_in[2];  // [64,1] fp32
  float* out  = (float*)d_out;               // [128,256,64] fp32
  float* Kmat = (float*)d_ws;                // TSTEPS*64*64 fp32 = 4 MB

  hipLaunchKernelGGL(kalman_gain_kernel, dim3(1), dim3(512), 0, stream,
                     Q, R, Kmat);
  hipLaunchKernelGGL(kalman_scan_kernel, dim3(BATCH / 16), dim3(128), 0, stream,
                     arr, Kmat, out);
}